// Decoder_61323543052777
// MI455X (gfx1250) — compile-verified
//
#include <hip/hip_runtime.h>
#include <hip/hip_bf16.h>
#include <stdint.h>

typedef __attribute__((ext_vector_type(2))) float v2f;
typedef __attribute__((ext_vector_type(4))) float v4f;
typedef __attribute__((ext_vector_type(8))) float v8f;

// One wave32 processes a tile of 16 edges with V_WMMA_F32_16X16X4_F32.
// Operand roles (vs round 1) are swapped to kill all B-masking VALU:
//   A (16x4  f32): W replicated across all 16 rows -> A[m][k] = W[map(k)].
//                  Per-lane layout (lanes 0-15: K=0,1 ; lanes 16-31: K=2,3)
//                  doesn't depend on m, so replication costs nothing.
//   B (4x16  f32): diff tile, B[k][n] = |row-col| of edge n at d=map(k).
//   D (16x16 f32): D[m][n] = dot(W_chunk, diff[n]) replicated over m;
//                  row M=0 => edge n's partial sits in lane n, acc VGPR 0.
// K-permutation trick: WMMA reduces over K, so per outer step t we assign
// d = 8t + 4*kh + {0..3} to the K slots of two chained WMMAs; every lane
// fetches one contiguous float4 (global_load_b128) from row, col and W.
__global__ __launch_bounds__(256) void edge_absdiff_dot_wmma(
    const float* __restrict__ X,        // [N,128]
    const int64_t* __restrict__ rIdx,   // [E]
    const int64_t* __restrict__ cIdx,   // [E]
    const float* __restrict__ W,        // [128]
    const float* __restrict__ bias,     // [1]
    float* __restrict__ out,            // [E]
    int E, int tiles)
{
    const int lane = threadIdx.x & 31;
    const int wave = blockIdx.x * (blockDim.x >> 5) + (threadIdx.x >> 5);
    if (wave >= tiles) return;          // wave-uniform exit keeps EXEC all-ones

    const int m  = lane & 15;           // edge slot in tile == N for B/C/D
    const int kh = lane >> 4;           // K-half selector
    const int e0 = wave * 16;

    int e = e0 + m;
    if (e >= E) e = E - 1;              // clamp (EXEC must stay all-ones for WMMA)

    const float* __restrict__ rowp = X + (size_t)rIdx[e] * 128;
    const float* __restrict__ colp = X + (size_t)cIdx[e] * 128;

    v8f acc0 = {0.f, 0.f, 0.f, 0.f, 0.f, 0.f, 0.f, 0.f};
    v8f acc1 = {0.f, 0.f, 0.f, 0.f, 0.f, 0.f, 0.f, 0.f};

#pragma unroll 4
    for (int t = 0; t < 16; ++t) {
        const int d = t * 8 + kh * 4;           // this lane's contiguous chunk
        v4f rv = *(const v4f*)(rowp + d);       // global_load_b128
        v4f cv = *(const v4f*)(colp + d);       // global_load_b128
        v4f wv = *(const v4f*)(W + d);          // uniform, cache-resident

        v4f diff;
        diff.x = __builtin_fabsf(rv.x - cv.x);
        diff.y = __builtin_fabsf(rv.y - cv.y);
        diff.z = __builtin_fabsf(rv.z - cv.z);
        diff.w = __builtin_fabsf(rv.w - cv.w);

        v2f a0 = {wv.x, wv.y};    v2f b0 = {diff.x, diff.y};
        v2f a1 = {wv.z, wv.w};    v2f b1 = {diff.z, diff.w};

        // 8 args: (neg_a, A, neg_b, B, c_mod, C, reuse_a, reuse_b)
        acc0 = __builtin_amdgcn_wmma_f32_16x16x4_f32(false, a0, false, b0,
                                                     (short)0, acc0, false, false);
        acc1 = __builtin_amdgcn_wmma_f32_16x16x4_f32(false, a1, false, b1,
                                                     (short)0, acc1, false, false);
    }

    // Row M=0 of D: lane n (n = 0..15) holds edge e0+n's dot in acc VGPR 0.
    // (Rows are replicated, so lanes 16-31 hold duplicates; only 0-15 store.)
    if (lane < 16) {
        const int ee = e0 + m;
        if (ee < E) out[ee] = acc0[0] + acc1[0] + bias[0];
    }
}

extern "C" void kernel_launch(void* const* d_in, const int* in_sizes, int n_in,
                              void* d_out, int out_size, void* d_ws, size_t ws_size,
                              hipStream_t stream) {
    const float*   X    = (const float*)d_in[0];     // inputs [N,128] f32
    const int64_t* rIdx = (const int64_t*)d_in[1];   // r_indices [E] i64
    const int64_t* cIdx = (const int64_t*)d_in[2];   // c_indices [E] i64
    const float*   W    = (const float*)d_in[3];     // W [128,1] f32
    const float*   b    = (const float*)d_in[4];     // b [1] f32
    float*         out  = (float*)d_out;             // [E,1] f32

    const int E = in_sizes[1];
    if (E <= 0) return;

    const int tiles         = (E + 15) / 16;
    const int wavesPerBlock = 8;                     // 256 threads = 8 wave32
    const int blocks        = (tiles + wavesPerBlock - 1) / wavesPerBlock;

    edge_absdiff_dot_wmma<<<blocks, wavesPerBlock * 32, 0, stream>>>(
        X, rIdx, cIdx, W, b, out, E, tiles);
}